// PointViewGenerator_81123342287422
// MI455X (gfx1250) — compile-verified
//
#include <hip/hip_runtime.h>
#include <cstdint>
#include <cfloat>

// ---------------------------------------------------------------------------
// Problem constants (from reference: B=256, N=65536, k=int(0.8*N)=52428)
// ---------------------------------------------------------------------------
#define BATCH   256
#define NPTS    65536
#define KSEL    52428
#define NBINS   1024           // histogram bins over float-bits of d^2
#define CHB     16             // histogram chunks per (view,batch)
#define CHC     64             // compaction chunks per (view,batch)
#define ITC     (NPTS / (CHC * 256))          // 4 iterations per compact block
#define CHR     16             // radius chunks per (view,batch)
#define CHUNKR  ((KSEL + CHR - 1) / CHR)
#define CHUNKW  2048           // points per rotate block
#define CHW     ((KSEL + CHUNKW - 1) / CHUNKW) // 26 rotate blocks per (view,batch)
#define ITW     (CHUNKW / 256) // 8 iterations per rotate block

typedef float v2f __attribute__((ext_vector_type(2)));
typedef float v8f __attribute__((ext_vector_type(8)));

// ordered-int encoding so unsigned atomicMin/Max implement float min/max
__device__ __forceinline__ unsigned ordEnc(float f) {
  unsigned u = __float_as_uint(f);
  return (u >> 31) ? ~u : (u | 0x80000000u);
}
__device__ __forceinline__ float ordDec(unsigned e) {
  unsigned u = (e >> 31) ? (e ^ 0x80000000u) : ~e;
  return __uint_as_float(u);
}
__device__ __forceinline__ unsigned wang(unsigned s) {
  s = (s ^ 61u) ^ (s >> 16); s *= 9u; s ^= s >> 4; s *= 0x27d4eb2du; s ^= s >> 15;
  return s;
}

// ---------------------------------------------------------------------------
// K0: zero/init workspace accumulators, gather crop centers pts[b, idx[b]]
// ---------------------------------------------------------------------------
__global__ void k_init(const float* __restrict__ pts, const int* __restrict__ idx1,
                       const int* __restrict__ idx2, unsigned* __restrict__ hist,
                       float* __restrict__ ctr, unsigned* __restrict__ bboxMin,
                       unsigned* __restrict__ bboxMax, unsigned* __restrict__ rad2) {
  int t = blockIdx.x * blockDim.x + threadIdx.x;
  if (t < 2 * BATCH * NBINS) hist[t] = 0u;
  if (t < 2 * BATCH * 3) { bboxMin[t] = 0xFFFFFFFFu; bboxMax[t] = 0u; }
  if (t < 2 * BATCH) {
    rad2[t] = 0u;
    int v = t / BATCH, b = t - v * BATCH;
    int ci = v ? idx2[b] : idx1[b];
    const float* p = pts + ((size_t)b * NPTS + (size_t)ci) * 3;
    ctr[t * 3 + 0] = p[0]; ctr[t * 3 + 1] = p[1]; ctr[t * 3 + 2] = p[2];
  }
}

// ---------------------------------------------------------------------------
// K1: per-(view,batch) histogram of d^2 float bits[30:21] (monotone binning)
// ---------------------------------------------------------------------------
__global__ void k_hist(const float* __restrict__ pts, const float* __restrict__ ctr,
                       unsigned* __restrict__ hist) {
  __shared__ unsigned lh[NBINS];
  for (int i = threadIdx.x; i < NBINS; i += blockDim.x) lh[i] = 0u;
  __syncthreads();
  int vb = blockIdx.x / CHB, chunk = blockIdx.x % CHB;
  int b = vb & (BATCH - 1);
  float cx = ctr[vb * 3 + 0], cy = ctr[vb * 3 + 1], cz = ctr[vb * 3 + 2];
  const float* base = pts + (size_t)b * NPTS * 3;
  int start = chunk * (NPTS / CHB);
  for (int i = threadIdx.x; i < NPTS / CHB; i += blockDim.x) {
    int p = start + i;
    __builtin_prefetch(base + (size_t)(p + 2 * 256) * 3, 0, 1);  // global_prefetch
    float x = base[p * 3 + 0] - cx;
    float y = base[p * 3 + 1] - cy;
    float z = base[p * 3 + 2] - cz;
    unsigned bin = __float_as_uint(x * x + y * y + z * z) >> 21;  // sign==0 -> <1024
    atomicAdd(&lh[bin], 1u);
  }
  __syncthreads();
  for (int i = threadIdx.x; i < NBINS; i += blockDim.x)
    if (lh[i]) atomicAdd(&hist[(size_t)vb * NBINS + i], lh[i]);
}

// ---------------------------------------------------------------------------
// K2: find threshold bin (bin containing the KSEL-th smallest d^2)
// ---------------------------------------------------------------------------
__global__ void k_thresh(const unsigned* __restrict__ hist, unsigned* __restrict__ thrBin) {
  int vb = blockIdx.x;
  if (threadIdx.x == 0) {
    const unsigned* h = hist + (size_t)vb * NBINS;
    unsigned cum = 0; unsigned t = NBINS - 1;
    for (int i = 0; i < NBINS; ++i) {
      unsigned c = h[i];
      if (cum + c >= (unsigned)KSEL) { t = (unsigned)i; break; }
      cum += c;
    }
    thrBin[vb] = t;
  }
}

// ---------------------------------------------------------------------------
// K3a: per-block counts of below-threshold ("low") and threshold-bin ("tie")
// ---------------------------------------------------------------------------
__global__ void k_count(const float* __restrict__ pts, const float* __restrict__ ctr,
                        const unsigned* __restrict__ thrBin,
                        unsigned* __restrict__ lowCnt, unsigned* __restrict__ tieCnt) {
  __shared__ unsigned sL[8], sT[8];
  int vb = blockIdx.x / CHC, blk = blockIdx.x % CHC;
  int b = vb & (BATCH - 1);
  float cx = ctr[vb * 3 + 0], cy = ctr[vb * 3 + 1], cz = ctr[vb * 3 + 2];
  unsigned t = thrBin[vb];
  const float* base = pts + (size_t)b * NPTS * 3;
  int start = blk * (NPTS / CHC);
  int lane = threadIdx.x & 31, wave = threadIdx.x >> 5;
  unsigned nlow = 0, ntie = 0;
  for (int it = 0; it < ITC; ++it) {
    int p = start + it * 256 + threadIdx.x;
    float x = base[p * 3 + 0] - cx;
    float y = base[p * 3 + 1] - cy;
    float z = base[p * 3 + 2] - cz;
    unsigned bin = __float_as_uint(x * x + y * y + z * z) >> 21;
    nlow += (bin < t) ? 1u : 0u;
    ntie += (bin == t) ? 1u : 0u;
  }
  for (int off = 16; off; off >>= 1) {
    nlow += __shfl_down(nlow, off);
    ntie += __shfl_down(ntie, off);
  }
  if (lane == 0) { sL[wave] = nlow; sT[wave] = ntie; }
  __syncthreads();
  if (threadIdx.x == 0) {
    unsigned a = 0, c = 0;
    for (int w = 0; w < 8; ++w) { a += sL[w]; c += sT[w]; }
    lowCnt[(size_t)vb * CHC + blk] = a;
    tieCnt[(size_t)vb * CHC + blk] = c;
  }
}

// ---------------------------------------------------------------------------
// K3b: exclusive scan of per-block counts -> deterministic write bases
// ---------------------------------------------------------------------------
__global__ void k_scan(const unsigned* __restrict__ lowCnt, const unsigned* __restrict__ tieCnt,
                       unsigned* __restrict__ lowBase, unsigned* __restrict__ tieBase,
                       unsigned* __restrict__ lowTotal, unsigned* __restrict__ needArr) {
  int vb = blockIdx.x;
  if (threadIdx.x == 0) {
    unsigned lo = 0, ti = 0;
    for (int i = 0; i < CHC; ++i) {
      lowBase[(size_t)vb * CHC + i] = lo;
      tieBase[(size_t)vb * CHC + i] = ti;
      lo += lowCnt[(size_t)vb * CHC + i];
      ti += tieCnt[(size_t)vb * CHC + i];
    }
    lowTotal[vb] = lo;
    needArr[vb] = ((unsigned)KSEL > lo) ? ((unsigned)KSEL - lo) : 0u;
  }
}

// ---------------------------------------------------------------------------
// K3c: deterministic compaction of selected raw points into view slots +
//      bbox min/max accumulation (ordered-int atomics, order-independent)
// ---------------------------------------------------------------------------
__global__ void k_write(const float* __restrict__ pts, const float* __restrict__ ctr,
                        const unsigned* __restrict__ thrBin,
                        const unsigned* __restrict__ lowBase, const unsigned* __restrict__ tieBase,
                        const unsigned* __restrict__ lowTotal, const unsigned* __restrict__ needArr,
                        float* __restrict__ dout,
                        unsigned* __restrict__ bboxMin, unsigned* __restrict__ bboxMax) {
  __shared__ unsigned wL[8], wT[8];
  __shared__ float sMn[8][3], sMx[8][3];
  int vb = blockIdx.x / CHC, blk = blockIdx.x % CHC;
  int b = vb & (BATCH - 1);
  float cx = ctr[vb * 3 + 0], cy = ctr[vb * 3 + 1], cz = ctr[vb * 3 + 2];
  unsigned t = thrBin[vb];
  unsigned lo = lowBase[(size_t)vb * CHC + blk];
  unsigned to = tieBase[(size_t)vb * CHC + blk];
  unsigned need = needArr[vb], below = lowTotal[vb];
  const float* base = pts + (size_t)b * NPTS * 3;
  float* dst = dout + 768 + (size_t)vb * KSEL * 3;
  int start = blk * (NPTS / CHC);
  int lane = threadIdx.x & 31, wave = threadIdx.x >> 5;
  unsigned lmask = (1u << lane) - 1u;
  float mnx = FLT_MAX, mny = FLT_MAX, mnz = FLT_MAX;
  float mxx = -FLT_MAX, mxy = -FLT_MAX, mxz = -FLT_MAX;
  for (int it = 0; it < ITC; ++it) {
    int p = start + it * 256 + threadIdx.x;
    float x0 = base[p * 3 + 0], y0 = base[p * 3 + 1], z0 = base[p * 3 + 2];
    float x = x0 - cx, y = y0 - cy, z = z0 - cz;
    unsigned bin = __float_as_uint(x * x + y * y + z * z) >> 21;
    bool low = bin < t, tie = bin == t;
    unsigned ml = (unsigned)__ballot(low);
    unsigned mt = (unsigned)__ballot(tie);
    if (lane == 0) { wL[wave] = __popc(ml); wT[wave] = __popc(mt); }
    __syncthreads();
    unsigned cl = 0, ct = 0, il = 0, itt = 0;
    for (int w = 0; w < 8; ++w) {
      unsigned a = wL[w], c = wT[w];
      if (w < wave) { cl += a; ct += c; }
      il += a; itt += c;
    }
    bool wr = false; unsigned pos = 0;
    if (low) {
      pos = lo + cl + __popc(ml & lmask);
      wr = true;
    } else if (tie) {
      unsigned r = to + ct + __popc(mt & lmask);
      if (r < need) { pos = below + r; wr = true; }
    }
    if (wr) {
      size_t o = (size_t)pos * 3;
      dst[o + 0] = x0; dst[o + 1] = y0; dst[o + 2] = z0;
      mnx = fminf(mnx, x0); mny = fminf(mny, y0); mnz = fminf(mnz, z0);
      mxx = fmaxf(mxx, x0); mxy = fmaxf(mxy, y0); mxz = fmaxf(mxz, z0);
    }
    lo += il; to += itt;
    __syncthreads();
  }
  for (int off = 16; off; off >>= 1) {
    mnx = fminf(mnx, __shfl_down(mnx, off)); mny = fminf(mny, __shfl_down(mny, off));
    mnz = fminf(mnz, __shfl_down(mnz, off)); mxx = fmaxf(mxx, __shfl_down(mxx, off));
    mxy = fmaxf(mxy, __shfl_down(mxy, off)); mxz = fmaxf(mxz, __shfl_down(mxz, off));
  }
  if (lane == 0) {
    sMn[wave][0] = mnx; sMn[wave][1] = mny; sMn[wave][2] = mnz;
    sMx[wave][0] = mxx; sMx[wave][1] = mxy; sMx[wave][2] = mxz;
  }
  __syncthreads();
  if (threadIdx.x == 0) {
    float a0 = sMn[0][0], a1 = sMn[0][1], a2 = sMn[0][2];
    float b0 = sMx[0][0], b1 = sMx[0][1], b2 = sMx[0][2];
    for (int w = 1; w < 8; ++w) {
      a0 = fminf(a0, sMn[w][0]); a1 = fminf(a1, sMn[w][1]); a2 = fminf(a2, sMn[w][2]);
      b0 = fmaxf(b0, sMx[w][0]); b1 = fmaxf(b1, sMx[w][1]); b2 = fmaxf(b2, sMx[w][2]);
    }
    atomicMin(&bboxMin[vb * 3 + 0], ordEnc(a0));
    atomicMin(&bboxMin[vb * 3 + 1], ordEnc(a1));
    atomicMin(&bboxMin[vb * 3 + 2], ordEnc(a2));
    atomicMax(&bboxMax[vb * 3 + 0], ordEnc(b0));
    atomicMax(&bboxMax[vb * 3 + 1], ordEnc(b1));
    atomicMax(&bboxMax[vb * 3 + 2], ordEnc(b2));
  }
}

// ---------------------------------------------------------------------------
// K4: bbox midpoint per (view,batch)
// ---------------------------------------------------------------------------
__global__ void k_center(const unsigned* __restrict__ bboxMin, const unsigned* __restrict__ bboxMax,
                         float* __restrict__ ocenter) {
  int t = blockIdx.x * blockDim.x + threadIdx.x;
  if (t < 2 * BATCH * 3) ocenter[t] = 0.5f * (ordDec(bboxMin[t]) + ordDec(bboxMax[t]));
}

// ---------------------------------------------------------------------------
// K5: max squared radius per (view,batch); non-negative floats -> bit atomicMax
// ---------------------------------------------------------------------------
__global__ void k_radius(const float* __restrict__ dout, const float* __restrict__ ocenter,
                         unsigned* __restrict__ rad2) {
  __shared__ float sM[8];
  int vb = blockIdx.x / CHR, chunk = blockIdx.x % CHR;
  float cx = ocenter[vb * 3 + 0], cy = ocenter[vb * 3 + 1], cz = ocenter[vb * 3 + 2];
  const float* src = dout + 768 + (size_t)vb * KSEL * 3;
  int start = chunk * CHUNKR;
  float m = 0.f;
  for (int i = start + threadIdx.x; i < start + CHUNKR && i < KSEL; i += blockDim.x) {
    float x = src[(size_t)i * 3 + 0] - cx;
    float y = src[(size_t)i * 3 + 1] - cy;
    float z = src[(size_t)i * 3 + 2] - cz;
    m = fmaxf(m, x * x + y * y + z * z);
  }
  for (int off = 16; off; off >>= 1) m = fmaxf(m, __shfl_down(m, off));
  int lane = threadIdx.x & 31, wave = threadIdx.x >> 5;
  if (lane == 0) sM[wave] = m;
  __syncthreads();
  if (threadIdx.x == 0) {
    for (int w = 1; w < 8; ++w) m = fmaxf(m, sM[w]);
    atomicMax(&rad2[vb], __float_as_uint(m));
  }
}

// ---------------------------------------------------------------------------
// K6: rotation params (1/r, cos, sin) + relative_center = c2 - c1
// ---------------------------------------------------------------------------
__global__ void k_final(const unsigned* __restrict__ rad2, const float* __restrict__ ocenter,
                        float* __restrict__ rotp, float* __restrict__ dout) {
  int t = blockIdx.x * blockDim.x + threadIdx.x;
  if (t < 2 * BATCH) {
    float r = sqrtf(__uint_as_float(rad2[t]));
    float rinv = (r > 0.f) ? (1.f / r) : 1.f;
    unsigned h = wang(0x9E3779B9u + (unsigned)t);
    float theta = (float)h * (6.283185307179586f / 4294967296.0f);
    rotp[t * 4 + 0] = rinv;
    rotp[t * 4 + 1] = cosf(theta);
    rotp[t * 4 + 2] = sinf(theta);
    rotp[t * 4 + 3] = 0.f;
  }
  if (t < BATCH * 3) dout[t] = ocenter[BATCH * 3 + t] - ocenter[t];
}

// ---------------------------------------------------------------------------
// K7: in-place normalize + rotate via V_WMMA_F32_16X16X4_F32.
// A (16x4) = R^T padded; B (4x16) = 16 points' (x,y,z,0) column-wise.
// D rows 0..2 = rotated coords: lane n (n<16) holds point n's (x,y,z) in
// D VGPRs 0..2. Two WMMAs per wave-iteration -> 32 points.
// ---------------------------------------------------------------------------
__global__ void k_rotate(float* __restrict__ dout, const float* __restrict__ ocenter,
                         const float* __restrict__ rotp) {
  int vb = blockIdx.x / CHW, chunk = blockIdx.x % CHW;
  float cx = ocenter[vb * 3 + 0], cy = ocenter[vb * 3 + 1], cz = ocenter[vb * 3 + 2];
  float rinv = rotp[vb * 4 + 0], c = rotp[vb * 4 + 1], s = rotp[vb * 4 + 2];
  float* base = dout + 768 + (size_t)vb * KSEL * 3;
  int lane = threadIdx.x & 31, wave = threadIdx.x >> 5;
  bool hi = lane >= 16;

  // A-matrix 16x4 layout: lane L<16 -> (A[L][0], A[L][1]); lane 16+L -> (A[L][2], A[L][3]).
  // A[e][d] = R[d][e]  (so D[e][n] = sum_d R[d][e] * q_n[d], matching einsum bnd,bde->bne)
  v2f a; a.x = 0.f; a.y = 0.f;
  if (lane == 0)       { a.x = c;   a.y = s; }   // (R00, R10) = (c,  s)
  else if (lane == 1)  { a.x = -s;  a.y = c; }   // (R01, R11) = (-s, c)
  else if (lane == 18) { a.x = 1.f; }            // A[2][2] = R22 = 1

  v8f zero8 = {0.f, 0.f, 0.f, 0.f, 0.f, 0.f, 0.f, 0.f};
  int start = chunk * CHUNKW;
  for (int it = 0; it < ITW; ++it) {
    int point = start + it * 256 + wave * 32 + lane;
    int idx = (point < KSEL) ? point : (KSEL - 1);   // keep EXEC all-ones for WMMA
    float px = base[(size_t)idx * 3 + 0];
    float py = base[(size_t)idx * 3 + 1];
    float pz = base[(size_t)idx * 3 + 2];
    float qx = (px - cx) * rinv, qy = (py - cy) * rinv, qz = (pz - cz) * rinv;

    // B-matrix 4x16 layout: lane n<16 -> (B[0][n],B[1][n]); lane 16+n -> (B[2][n],B[3][n])
    float zlo = __shfl(qz, lane & 15);  // z of point (pbase + n) for high half
    float xhi = __shfl(qx, lane | 16);  // x of point (pbase + 16 + n) for low half
    float yhi = __shfl(qy, lane | 16);
    v2f b1, b2;
    b1.x = hi ? zlo : qx;  b1.y = hi ? 0.f : qy;   // points pbase..pbase+15
    b2.x = hi ? qz  : xhi; b2.y = hi ? 0.f : yhi;  // points pbase+16..pbase+31

    v8f d1 = __builtin_amdgcn_wmma_f32_16x16x4_f32(false, a, false, b1,
                                                   (short)0, zero8, false, false);
    v8f d2 = __builtin_amdgcn_wmma_f32_16x16x4_f32(false, a, false, b2,
                                                   (short)0, zero8, false, false);

    // move second WMMA's results (lanes 0..15) up to lanes 16..31 for coalesced stores
    float sx = __shfl(d2[0], lane & 15);
    float sy = __shfl(d2[1], lane & 15);
    float sz = __shfl(d2[2], lane & 15);
    float ox = hi ? sx : d1[0];
    float oy = hi ? sy : d1[1];
    float oz = hi ? sz : d1[2];
    if (point < KSEL) {
      base[(size_t)point * 3 + 0] = ox;
      base[(size_t)point * 3 + 1] = oy;
      base[(size_t)point * 3 + 2] = oz;
    }
  }
}

// ---------------------------------------------------------------------------
// Host launcher
// ---------------------------------------------------------------------------
extern "C" void kernel_launch(void* const* d_in, const int* in_sizes, int n_in,
                              void* d_out, int out_size, void* d_ws, size_t ws_size,
                              hipStream_t stream) {
  const float* pts = (const float*)d_in[0];
  const int* idx1 = (const int*)d_in[1];
  const int* idx2 = (const int*)d_in[2];
  float* dout = (float*)d_out;

  // workspace layout (~2.7 MB total)
  unsigned* hist     = (unsigned*)d_ws;                          // 2*B*NBINS
  float*    ctr      = (float*)(hist + 2 * BATCH * NBINS);       // 2*B*3
  unsigned* thrBin   = (unsigned*)(ctr + 2 * BATCH * 3);         // 2*B
  unsigned* lowCnt   = thrBin + 2 * BATCH;                       // 2*B*CHC
  unsigned* tieCnt   = lowCnt + 2 * BATCH * CHC;                 // 2*B*CHC
  unsigned* lowBase  = tieCnt + 2 * BATCH * CHC;                 // 2*B*CHC
  unsigned* tieBase  = lowBase + 2 * BATCH * CHC;                // 2*B*CHC
  unsigned* lowTotal = tieBase + 2 * BATCH * CHC;                // 2*B
  unsigned* needArr  = lowTotal + 2 * BATCH;                     // 2*B
  unsigned* bboxMin  = needArr + 2 * BATCH;                      // 2*B*3
  unsigned* bboxMax  = bboxMin + 2 * BATCH * 3;                  // 2*B*3
  unsigned* rad2     = bboxMax + 2 * BATCH * 3;                  // 2*B
  float*    ocenter  = (float*)(rad2 + 2 * BATCH);               // 2*B*3
  float*    rotp     = ocenter + 2 * BATCH * 3;                  // 2*B*4

  k_init<<<(2 * BATCH * NBINS + 255) / 256, 256, 0, stream>>>(pts, idx1, idx2, hist, ctr,
                                                              bboxMin, bboxMax, rad2);
  k_hist<<<2 * BATCH * CHB, 256, 0, stream>>>(pts, ctr, hist);
  k_thresh<<<2 * BATCH, 32, 0, stream>>>(hist, thrBin);
  k_count<<<2 * BATCH * CHC, 256, 0, stream>>>(pts, ctr, thrBin, lowCnt, tieCnt);
  k_scan<<<2 * BATCH, 32, 0, stream>>>(lowCnt, tieCnt, lowBase, tieBase, lowTotal, needArr);
  k_write<<<2 * BATCH * CHC, 256, 0, stream>>>(pts, ctr, thrBin, lowBase, tieBase, lowTotal,
                                               needArr, dout, bboxMin, bboxMax);
  k_center<<<(2 * BATCH * 3 + 255) / 256, 256, 0, stream>>>(bboxMin, bboxMax, ocenter);
  k_radius<<<2 * BATCH * CHR, 256, 0, stream>>>(dout, ocenter, rad2);
  k_final<<<(BATCH * 3 + 255) / 256, 256, 0, stream>>>(rad2, ocenter, rotp, dout);
  k_rotate<<<2 * BATCH * CHW, 256, 0, stream>>>(dout, ocenter, rotp);
}